// MolecularProdRuleEmbeddingLastLayer_5076651344548
// MI455X (gfx1250) — compile-verified
//
#include <hip/hip_runtime.h>
#include <stdint.h>

// ---------------------------------------------------------------------------
// MolecularProdRuleEmbedding — CDNA5 (gfx1250) implementation
//   Phase 1: per-rule 3-layer GNN, GEMMs via V_WMMA_F32_16X16X4_F32
//   Phase 2: bandwidth-bound index gather, b128 non-temporal stores
// ---------------------------------------------------------------------------

#define R_RULES 4096
#define NMAX    24
#define EMAX    32
#define DIM     64
#define OUTD    64
#define LAYERS  3
#define BT      (512 * 1024)

#define RPW     2          // rules (waves) per workgroup
#define STRIDE  65         // padded LDS row stride (floats) -> conflict-free column reads

typedef float v2f __attribute__((ext_vector_type(2)));
typedef float v4f __attribute__((ext_vector_type(4)));
typedef float v8f __attribute__((ext_vector_type(8)));

// act_in : [32][STRIDE] LDS, rows = M, cols = K (64 valid)
// Wl     : [64][64] LDS row-major (row = K/d, col = N/f)
// bl     : [64] bias
// out    : [32][STRIDE] LDS (may alias act_in; written only after all K-steps)
__device__ __forceinline__ void gemm64_wmma(const float* __restrict__ act_in,
                                            float* __restrict__ out_buf,
                                            const float* __restrict__ Wl,
                                            const float* __restrict__ bl,
                                            int lane, bool do_relu)
{
    const int l16  = lane & 15;
    const int half = lane >> 4;          // 0: lanes 0-15, 1: lanes 16-31

    v8f acc[2][4];
    for (int mt = 0; mt < 2; ++mt)
        for (int nt = 0; nt < 4; ++nt) {
            float bv = bl[nt * 16 + l16];          // C(m,n) = bias[n]
            v8f c;
            for (int v = 0; v < 8; ++v) c[v] = bv;
            acc[mt][nt] = c;
        }

    for (int k0 = 0; k0 < DIM; k0 += 4) {
        const int kb = k0 + (half << 1);           // K lane-pair base per ISA A/B layout

        v2f a0, a1;
        a0.x = act_in[(l16)      * STRIDE + kb];
        a0.y = act_in[(l16)      * STRIDE + kb + 1];
        a1.x = act_in[(16 + l16) * STRIDE + kb];
        a1.y = act_in[(16 + l16) * STRIDE + kb + 1];

        v2f bm[4];
        for (int nt = 0; nt < 4; ++nt) {
            bm[nt].x = Wl[kb * DIM + nt * 16 + l16];
            bm[nt].y = Wl[(kb + 1) * DIM + nt * 16 + l16];
        }

        for (int nt = 0; nt < 4; ++nt) {
            acc[0][nt] = __builtin_amdgcn_wmma_f32_16x16x4_f32(
                false, a0, false, bm[nt], (short)0, acc[0][nt], false, false);
            acc[1][nt] = __builtin_amdgcn_wmma_f32_16x16x4_f32(
                false, a1, false, bm[nt], (short)0, acc[1][nt], false, false);
        }
    }

    // ReLU + writeback (C/D layout: VGPR v -> row v (+8 for upper lane half))
    for (int mt = 0; mt < 2; ++mt)
        for (int nt = 0; nt < 4; ++nt)
            for (int v = 0; v < 8; ++v) {
                float x = acc[mt][nt][v];
                if (do_relu) x = fmaxf(x, 0.0f);
                int row = mt * 16 + v + half * 8;
                int col = nt * 16 + l16;
                out_buf[row * STRIDE + col] = x;
            }
}

__global__ __launch_bounds__(32 * RPW)
void gnn_rules_kernel(const float* __restrict__ atom_embed,   // [20,64]
                      const float* __restrict__ bond_embed,   // [40,64]
                      const float* __restrict__ W_l2l,        // [3,64,64]
                      const float* __restrict__ b_l2l,        // [3,64]
                      const float* __restrict__ W_out,        // [64,64]
                      const float* __restrict__ b_out,        // [64]
                      const int*   __restrict__ node_sym,     // [R,NMAX]
                      const int*   __restrict__ edge_sym,     // [R,EMAX]
                      const int*   __restrict__ edge_nodes,   // [R,EMAX,2]
                      const unsigned char* __restrict__ edge_mask, // [R,EMAX] bool
                      const int*   __restrict__ last_node,    // [R]
                      float*       __restrict__ out_rule)     // [R,64]
{
    __shared__ float sW[LAYERS * DIM * DIM];   // 48 KB
    __shared__ float sWo[DIM * DIM];           // 16 KB
    __shared__ float sBias[LAYERS * DIM];
    __shared__ float sBo[DIM];
    __shared__ float sE [RPW][32 * STRIDE];    // edge feats (becomes e_mut, then new e)
    __shared__ float sN [RPW][32 * STRIDE];    // node feats
    __shared__ float sNm[RPW][32 * STRIDE];    // n_mut
    __shared__ int   sEN[RPW][EMAX * 2];
    __shared__ float sMask[RPW][EMAX];

    const int tid  = threadIdx.x;
    const int wave = tid >> 5;
    const int lane = tid & 31;
    const int rule = blockIdx.x * RPW + wave;

    // ---- stage weights (block-cooperative) ----
    for (int i = tid; i < LAYERS * DIM * DIM; i += blockDim.x) sW[i]   = W_l2l[i];
    for (int i = tid; i < DIM * DIM;          i += blockDim.x) sWo[i]  = W_out[i];
    for (int i = tid; i < LAYERS * DIM;       i += blockDim.x) sBias[i] = b_l2l[i];
    for (int i = tid; i < DIM;                i += blockDim.x) sBo[i]  = b_out[i];

    // ---- per-rule metadata (wave-private) ----
    sEN[wave][lane * 2 + 0] = edge_nodes[(rule * EMAX + lane) * 2 + 0];
    sEN[wave][lane * 2 + 1] = edge_nodes[(rule * EMAX + lane) * 2 + 1];
    sMask[wave][lane]       = edge_mask[rule * EMAX + lane] ? 1.0f : 0.0f;

    // ---- n = bond_embed[node_sym]; zero-pad rows NMAX..31 ----
    for (int row = 0; row < 32; ++row) {
        float v0 = 0.0f, v1 = 0.0f;
        if (row < NMAX) {
            int s = node_sym[rule * NMAX + row];
            v0 = bond_embed[s * DIM + lane];
            v1 = bond_embed[s * DIM + lane + 32];
        }
        sN[wave][row * STRIDE + lane]      = v0;
        sN[wave][row * STRIDE + lane + 32] = v1;
    }
    // ---- e = atom_embed[edge_sym] ----
    for (int row = 0; row < EMAX; ++row) {
        int s = edge_sym[rule * EMAX + row];
        sE[wave][row * STRIDE + lane]      = atom_embed[s * DIM + lane];
        sE[wave][row * STRIDE + lane + 32] = atom_embed[s * DIM + lane + 32];
    }
    __syncthreads();   // weights visible to all waves

    float* eBuf  = sE[wave];
    float* nBuf  = sN[wave];
    float* nmBuf = sNm[wave];

    for (int l = 0; l < LAYERS; ++l) {
        const float* Wl = &sW[l * DIM * DIM];
        const float* bl = &sBias[l * DIM];

        // 1) e_mut = e + n[a] + n[b]   (all edges, in place)
        for (int i = 0; i < EMAX; ++i) {
            int a = sEN[wave][i * 2 + 0];
            int b = sEN[wave][i * 2 + 1];
            eBuf[i * STRIDE + lane]      += nBuf[a * STRIDE + lane]      + nBuf[b * STRIDE + lane];
            eBuf[i * STRIDE + lane + 32] += nBuf[a * STRIDE + lane + 32] + nBuf[b * STRIDE + lane + 32];
        }

        // 2) n_mut = n + scatter-add of masked e_mut (wave-sequential, no atomics needed)
        for (int row = 0; row < 32; ++row) {
            nmBuf[row * STRIDE + lane]      = nBuf[row * STRIDE + lane];
            nmBuf[row * STRIDE + lane + 32] = nBuf[row * STRIDE + lane + 32];
        }
        for (int i = 0; i < EMAX; ++i) {
            if (sMask[wave][i] != 0.0f) {
                int a = sEN[wave][i * 2 + 0];
                int b = sEN[wave][i * 2 + 1];
                float c0 = eBuf[i * STRIDE + lane];
                float c1 = eBuf[i * STRIDE + lane + 32];
                nmBuf[a * STRIDE + lane]      += c0;
                nmBuf[a * STRIDE + lane + 32] += c1;
                nmBuf[b * STRIDE + lane]      += c0;
                nmBuf[b * STRIDE + lane + 32] += c1;
            }
        }

        // 3) e = relu(e_mut @ W + b)   (in place, safe: reg-accumulated)
        gemm64_wmma(eBuf, eBuf, Wl, bl, lane, true);
        // 4) n = relu(n_mut @ W + b)   (nmBuf preserved for v_last)
        gemm64_wmma(nmBuf, nBuf, Wl, bl, lane, true);
    }

    // ---- head: out_rule = tanh(n_mut[last] @ W_out + b_out) ----
    int ln = last_node[rule];
    float o0 = sBo[lane], o1 = sBo[lane + 32];
    for (int d = 0; d < DIM; ++d) {
        float v = nmBuf[ln * STRIDE + d];          // LDS broadcast
        o0 += v * sWo[d * DIM + lane];
        o1 += v * sWo[d * DIM + lane + 32];
    }
    out_rule[rule * OUTD + lane]      = tanhf(o0);
    out_rule[rule * OUTD + lane + 32] = tanhf(o1);
}

// ---------------------------------------------------------------------------
// Phase 2: out[b,t,:] = idx < R ? out_rule[clip(idx)] : 0     (134 MB stream)
// ---------------------------------------------------------------------------
__global__ __launch_bounds__(256)
void gather_kernel(const float* __restrict__ out_rule,
                   const int*   __restrict__ prod_rule_idx,
                   float*       __restrict__ out)
{
    int gid = blockIdx.x * blockDim.x + threadIdx.x;   // one 4-float chunk per thread
    if (gid >= BT * 16) return;
    int bt    = gid >> 4;
    int chunk = gid & 15;
    int idx   = prod_rule_idx[bt];
    v4f val = (v4f){0.f, 0.f, 0.f, 0.f};
    if (idx < R_RULES) {
        int safe = idx < 0 ? 0 : idx;
        val = ((const v4f*)(out_rule + (size_t)safe * OUTD))[chunk];
    }
    __builtin_nontemporal_store(val, (v4f*)out + gid);
}

extern "C" void kernel_launch(void* const* d_in, const int* in_sizes, int n_in,
                              void* d_out, int out_size, void* d_ws, size_t ws_size,
                              hipStream_t stream)
{
    const float* atom_embed = (const float*)d_in[0];
    const float* bond_embed = (const float*)d_in[1];
    const float* W_l2l      = (const float*)d_in[2];
    const float* b_l2l      = (const float*)d_in[3];
    const float* W_out      = (const float*)d_in[4];
    const float* b_out      = (const float*)d_in[5];
    const int*   node_sym   = (const int*)d_in[6];
    const int*   edge_sym   = (const int*)d_in[7];
    const int*   edge_nodes = (const int*)d_in[8];
    const unsigned char* edge_mask = (const unsigned char*)d_in[9];
    const int*   last_node  = (const int*)d_in[10];
    const int*   prod_rule_idx = (const int*)d_in[11];

    float* out      = (float*)d_out;
    float* out_rule = (float*)d_ws;                 // R*64 floats scratch

    dim3 g1(R_RULES / RPW), b1(32 * RPW);
    gnn_rules_kernel<<<g1, b1, 0, stream>>>(atom_embed, bond_embed, W_l2l, b_l2l,
                                            W_out, b_out, node_sym, edge_sym,
                                            edge_nodes, edge_mask, last_node, out_rule);

    const int total = BT * 16;
    dim3 g2((total + 255) / 256), b2(256);
    gather_kernel<<<g2, b2, 0, stream>>>(out_rule, prod_rule_idx, out);
}